// JetMoeTopKGating_25546465477251
// MI455X (gfx1250) — compile-verified
//
#include <hip/hip_runtime.h>
#include <hip/hip_bf16.h>

typedef __attribute__((ext_vector_type(2))) float v2f;
typedef __attribute__((ext_vector_type(8))) float v8f;

// Problem constants (from reference)
constexpr int N_TOK = 131072;
constexpr int D     = 2048;
constexpr int E     = 8;
constexpr int TOPK  = 2;
constexpr int NK    = N_TOK * TOPK;          // 262144 flat slots
constexpr int EXP_OFF    = 3 * NK;           // expert_size offset in d_out
constexpr int LOGITS_OFF = 3 * NK + E;       // logits offset in d_out

// GEMV tiling
constexpr int BM = 128;        // tokens per block (8 waves x 16)
constexpr int BK = 64;         // K chunk
constexpr int HPAD = BK + 4;   // LDS row stride in floats (bank-conflict-free A reads)
constexpr int NCH = D / BK;    // 32 chunks

// Sort blocking: 512 flat slots per block
constexpr int SBLK  = 512;
constexpr int NSBLK = NK / SBLK;             // 512 blocks

// Low 32 bits of a generic pointer into __shared__ == the LDS (addrspace-3)
// byte address (AMDGPU generic LDS pointers are {shared_base_hi32, lds_off32}).
__device__ __forceinline__ unsigned lds_off(const void* p) {
    return (unsigned)(unsigned long long)p;
}

// ---------------------------------------------------------------------------
// Kernel 1: logits = H @ W^T via V_WMMA_F32_16X16X4_F32.
// Double-buffered LDS tiles filled with GLOBAL_LOAD_ASYNC_TO_LDS (ASYNCcnt),
// so DMA of chunk c+1 overlaps WMMA on chunk c.
// ---------------------------------------------------------------------------
__global__ __launch_bounds__(256) void
logits_wmma_kernel(const float* __restrict__ H, const float* __restrict__ W,
                   float* __restrict__ logits) {
    __shared__ float bufH[2][BM][HPAD];   // 2 x ~34 KB
    __shared__ float bufW[2][BK][16];     // 2 x 4 KB, K-major, cols 8..15 == 0

    const int t    = threadIdx.x;
    const int lane = t & 31;
    const int wave = t >> 5;
    const int hi   = lane >> 4;     // 0: K pair {0,1}; 1: K pair {2,3}
    const int m    = lane & 15;     // token-in-tile for A, column N for B/C
    const int blockBase = blockIdx.x * BM;

    // Zero both W buffers (pad columns 8..15 must read as 0 forever).
    {
        float* wf = &bufW[0][0][0];
        #pragma unroll
        for (int i = 0; i < (2 * BK * 16) / 256; ++i) wf[t + i * 256] = 0.0f;
    }
    __syncthreads();   // zero-init visible before async writes land

    // Issue one chunk's worth of async global->LDS copies (10 per wave:
    // 8 x b128 for H, 2 x b32 for W).
    auto issue_chunk = [&](int k0, int sel) {
        #pragma unroll
        for (int i = 0; i < (BM * BK / 4) / 256; ++i) {     // 8 iters
            int f   = t + i * 256;
            int tok = f >> 4;
            int kk4 = f & 15;
            unsigned l = lds_off(&bufH[sel][tok][kk4 * 4]);
            const float* g = H + (size_t)(blockBase + tok) * D + k0 + kk4 * 4;
            asm volatile("global_load_async_to_lds_b128 %0, %1, off th:TH_LOAD_NT"
                         :: "v"(l), "v"(g) : "memory");
        }
        #pragma unroll
        for (int i = 0; i < (E * BK) / 256; ++i) {          // 2 iters
            int idx = t + i * 256;
            int e = idx >> 6;
            int k = idx & 63;
            unsigned l = lds_off(&bufW[sel][k][e]);
            const float* g = W + (size_t)e * D + k0 + k;
            asm volatile("global_load_async_to_lds_b32 %0, %1, off"
                         :: "v"(l), "v"(g) : "memory");
        }
    };

    v8f acc = {0.f, 0.f, 0.f, 0.f, 0.f, 0.f, 0.f, 0.f};

    issue_chunk(0, 0);

    for (int c = 0; c < NCH; ++c) {
        const int cur = c & 1;
        if (c + 1 < NCH) {
            issue_chunk((c + 1) * BK, cur ^ 1);
            // async ops complete in order: <=10 outstanding => chunk c landed
            asm volatile("s_wait_asynccnt 0xa" ::: "memory");
        } else {
            asm volatile("s_wait_asynccnt 0x0" ::: "memory");
        }
        __syncthreads();   // every wave waited its own copies => tile complete

        // Gather all fragments first, then back-to-back WMMAs (one dscnt wait).
        v2f af[16], bf[16];
        #pragma unroll
        for (int q = 0; q < 16; ++q) {
            const int ka = q * 4 + 2 * hi;
            af[q][0] = bufH[cur][wave * 16 + m][ka];
            af[q][1] = bufH[cur][wave * 16 + m][ka + 1];
            bf[q][0] = bufW[cur][ka][m];
            bf[q][1] = bufW[cur][ka + 1][m];
        }
        #pragma unroll
        for (int q = 0; q < 16; ++q) {
            acc = __builtin_amdgcn_wmma_f32_16x16x4_f32(
                /*neg_a=*/false, af[q], /*neg_b=*/false, bf[q],
                /*c_mod=*/(short)0, acc, /*reuse_a=*/false, /*reuse_b=*/false);
        }
        __syncthreads();   // done reading buf[cur]; next iter may DMA into it
    }

    // C/D layout: VGPR r, lane: M = r + 8*hi, N = lane&15. Keep N<8.
    if (m < E) {
        #pragma unroll
        for (int r = 0; r < 8; ++r) {
            int tok = blockBase + wave * 16 + r + 8 * hi;
            logits[(size_t)tok * E + m] = acc[r];
        }
    }
}

// ---------------------------------------------------------------------------
// Kernel 2: per-token top-2 + softmax(2) + per-block expert histogram.
// Block = 256 tokens = 512 flat slots (matches sort blocking).
// ---------------------------------------------------------------------------
__global__ __launch_bounds__(256) void
topk_kernel(const float* __restrict__ logits, int* __restrict__ flatExperts,
            float* __restrict__ gates, int* __restrict__ blockCounts) {
    __shared__ int cnt[E];
    const int t = threadIdx.x;
    if (t < E) cnt[t] = 0;
    __syncthreads();

    const int tok = blockIdx.x * 256 + t;
    const float4 a = ((const float4*)(logits + (size_t)tok * E))[0];
    const float4 c = ((const float4*)(logits + (size_t)tok * E))[1];
    float l[E] = {a.x, a.y, a.z, a.w, c.x, c.y, c.z, c.w};

    // Strict '>' keeps the lowest index on ties (matches jax.lax.top_k).
    int i0 = 0; float v0 = l[0];
    #pragma unroll
    for (int e = 1; e < E; ++e) if (l[e] > v0) { v0 = l[e]; i0 = e; }
    int i1 = (i0 == 0) ? 1 : 0; float v1 = l[i1];
    #pragma unroll
    for (int e = 0; e < E; ++e)
        if (e != i0 && l[e] > v1) { v1 = l[e]; i1 = e; }

    // softmax over {v0, v1}, v0 >= v1.
    float e1 = __expf(v1 - v0);
    float inv = 1.0f / (1.0f + e1);
    float g0 = inv;
    float g1 = e1 * inv;

    flatExperts[2 * tok]     = i0;
    flatExperts[2 * tok + 1] = i1;
    gates[2 * tok]     = g0;
    gates[2 * tok + 1] = g1;

    atomicAdd(&cnt[i0], 1);
    atomicAdd(&cnt[i1], 1);
    __syncthreads();
    if (t < E) blockCounts[blockIdx.x * E + t] = cnt[t];
}

// ---------------------------------------------------------------------------
// Kernel 3: single block. Per-expert exclusive scan over blocks + expert bases.
// ---------------------------------------------------------------------------
__global__ __launch_bounds__(256) void
scan_kernel(const int* __restrict__ blockCounts, int* __restrict__ blockOffsets,
            float* __restrict__ expert_size_out) {
    __shared__ int totals[E];
    __shared__ int base[E];
    const int t = threadIdx.x;
    if (t < E) {
        int run = 0;
        for (int b = 0; b < NSBLK; ++b) {
            blockOffsets[b * E + t] = run;
            run += blockCounts[b * E + t];
        }
        totals[t] = run;
    }
    __syncthreads();
    if (t == 0) {
        int run = 0;
        for (int e = 0; e < E; ++e) {
            base[e] = run;
            expert_size_out[e] = (float)totals[e];
            run += totals[e];
        }
    }
    __syncthreads();
    for (int i = t; i < NSBLK * E; i += 256)
        blockOffsets[i] += base[i & (E - 1)];
}

// ---------------------------------------------------------------------------
// Kernel 4: stable scatter. Rank within block computed from LDS (deterministic).
// ---------------------------------------------------------------------------
__global__ __launch_bounds__(256) void
scatter_kernel(const int* __restrict__ flatExperts, const float* __restrict__ gates,
               const int* __restrict__ blockOffsets, float* __restrict__ out) {
    __shared__ int ex[SBLK];
    const int b = blockIdx.x;
    const int t = threadIdx.x;
    const int baseFlat = b * SBLK;
    ex[t]       = flatExperts[baseFlat + t];
    ex[t + 256] = flatExperts[baseFlat + t + 256];
    __syncthreads();

    for (int s = t; s < SBLK; s += 256) {
        const int e = ex[s];
        int rank = 0;
        for (int j = 0; j < s; ++j) rank += (ex[j] == e) ? 1 : 0;
        const int dest = blockOffsets[b * E + e] + rank;
        const int flatIdx = baseFlat + s;
        out[dest]          = (float)flatIdx;          // index_sorted_experts
        out[NK + dest]     = (float)(flatIdx >> 1);   // batch_index
        out[2 * NK + dest] = gates[flatIdx];          // batch_gates
    }
}

extern "C" void kernel_launch(void* const* d_in, const int* in_sizes, int n_in,
                              void* d_out, int out_size, void* d_ws, size_t ws_size,
                              hipStream_t stream) {
    const float* H = (const float*)d_in[0];   // [131072, 2048] f32
    const float* W = (const float*)d_in[1];   // [8, 2048] f32
    float* out = (float*)d_out;
    float* logits = out + LOGITS_OFF;

    char* ws = (char*)d_ws;
    int*   flatExperts  = (int*)ws;                          // NK ints
    float* gates        = (float*)(ws + (size_t)NK * 4);     // NK floats
    int*   blockCounts  = (int*)(ws + (size_t)2 * NK * 4);   // NSBLK*E ints
    int*   blockOffsets = blockCounts + NSBLK * E;           // NSBLK*E ints

    logits_wmma_kernel<<<N_TOK / BM, 256, 0, stream>>>(H, W, logits);
    topk_kernel<<<N_TOK / 256, 256, 0, stream>>>(logits, flatExperts, gates, blockCounts);
    scan_kernel<<<1, 256, 0, stream>>>(blockCounts, blockOffsets, out + EXP_OFF);
    scatter_kernel<<<NSBLK, 256, 0, stream>>>(flatExperts, gates, blockOffsets, out);
}